// MasterFormula_60155311948095
// MI455X (gfx1250) — compile-verified
//
#include <hip/hip_runtime.h>
#include <math.h>

// ---------------------------------------------------------------------------
// Problem constants (from reference): BATCH=131072, K=64, DEPTH=6
//   base  : (B, 65) complex128   (col 0 == 1, cols 1..64 == x)
//   Y     = base @ W_all^T, W_all = 126 real softmax rows of length 65
//           cols 0..63  : leaf level
//           cols 64..125: internal rows, consumed level 4 .. level 0
//   tree  : 63 x eml(a,b) = exp(log(a)*log(b)) complex combines (f64)
// ---------------------------------------------------------------------------
#define BATCH   131072
#define KDIM    64
#define KB      65          // k+1 terminals
#define KP      68          // K padded to multiple of 4 for WMMA
#define NC      128         // 126 weight rows padded to 128 columns
#define ROWS    64          // batch rows per block
#define THREADS 64          // 2 waves (wave32)

typedef float  v2f __attribute__((ext_vector_type(2)));
typedef float  v8f __attribute__((ext_vector_type(8)));

// ---------------------------------------------------------------------------
// Kernel 1: softmax of all weight rows (f64 math), packed to f32 W[NC][KP]
// plus the 62 f64 child scalars c[j] (= softmax weight of the child terminal).
// ---------------------------------------------------------------------------
__global__ __launch_bounds__(128)
void eml_weights_kernel(const double* __restrict__ internal_logits, // (62, 66)
                        const double* __restrict__ leaf_logits,     // (64, 65)
                        float* __restrict__ W,                      // (NC, KP)
                        double* __restrict__ cvals)                 // (62)
{
    int j = threadIdx.x;
    if (j < 64) {
        const double* row = leaf_logits + (size_t)j * KB;
        double mx = row[0];
        for (int k = 1; k < KB; ++k) mx = fmax(mx, row[k]);
        double s = 0.0;
        for (int k = 0; k < KB; ++k) s += exp(row[k] - mx);
        double inv = 1.0 / s;
        for (int k = 0; k < KP; ++k)
            W[j * KP + k] = (k < KB) ? (float)(exp(row[k] - mx) * inv) : 0.0f;
    } else if (j < 126) {
        int r = j - 64;
        const double* row = internal_logits + (size_t)r * (KB + 1); // 66 logits
        double mx = row[0];
        for (int k = 1; k < KB + 1; ++k) mx = fmax(mx, row[k]);
        double s = 0.0;
        for (int k = 0; k < KB + 1; ++k) s += exp(row[k] - mx);
        double inv = 1.0 / s;
        for (int k = 0; k < KP; ++k)
            W[j * KP + k] = (k < KB) ? (float)(exp(row[k] - mx) * inv) : 0.0f;
        cvals[r] = exp(row[KB] - mx) * inv;   // child-terminal weight (f64)
    } else if (j < 128) {
        for (int k = 0; k < KP; ++k) W[j * KP + k] = 0.0f;  // pad cols 126..127
    }
}

// ---------------------------------------------------------------------------
// eml(a,b) = exp(log(a) * log(b)) for complex doubles (principal branch).
// ---------------------------------------------------------------------------
__device__ __forceinline__
void eml_c(double ar, double ai, double br, double bi, double& rr, double& ri)
{
    double lar = 0.5 * log(ar * ar + ai * ai);
    double laa = atan2(ai, ar);
    double lbr = 0.5 * log(br * br + bi * bi);
    double lba = atan2(bi, br);
    double pr  = lar * lbr - laa * lba;
    double pi_ = lar * lba + laa * lbr;
    double m   = exp(pr);
    double s, c;
    sincos(pi_, &s, &c);
    rr = m * c;
    ri = m * s;
}

// ---------------------------------------------------------------------------
// Kernel 2: fused  Y = base @ W^T  (f32 WMMA, re & im as two real GEMMs)
//           + f64 eml tree per row.
// ---------------------------------------------------------------------------
__global__ __launch_bounds__(THREADS)
void eml_fused_kernel(const double* __restrict__ x,   // (B, 64) complex interleaved
                      const float*  __restrict__ Wg,  // (NC, KP)
                      const double* __restrict__ cg,  // (62)
                      double*       __restrict__ out) // (B) complex interleaved
{
    __shared__ float  Xre[ROWS][KP];
    __shared__ float  Xim[ROWS][KP];
    __shared__ float  Wl [NC ][KP];
    __shared__ float  Yre[ROWS][NC];
    __shared__ float  Yim[ROWS][NC];
    __shared__ double cl[64];

    const int t  = threadIdx.x;
    const int r0 = blockIdx.x * ROWS;

    // ---- stage weights ----
    for (int i = t; i < NC * KP; i += THREADS) (&Wl[0][0])[i] = Wg[i];
    if (t < 62) cl[t] = cg[t];

    // ---- stage x tile (coalesced f64 loads, convert to f32) ----
    // each row of x = 128 doubles (64 complex, interleaved re/im)
    for (int i = t; i < ROWS * 128; i += THREADS) {
        double v  = x[(size_t)r0 * 128 + i];
        int row   = i >> 7;
        int rem   = i & 127;
        int kk    = rem >> 1;           // complex element 0..63 -> base col kk+1
        if (rem & 1) Xim[row][kk + 1] = (float)v;
        else         Xre[row][kk + 1] = (float)v;
    }
    for (int i = t; i < ROWS; i += THREADS) {
        Xre[i][0] = 1.0f; Xim[i][0] = 0.0f;       // base column 0 == 1+0i
        Xre[i][65] = Xre[i][66] = Xre[i][67] = 0.0f;
        Xim[i][65] = Xim[i][66] = Xim[i][67] = 0.0f;
    }
    __syncthreads();

    // ---- GEMM phase: v_wmma_f32_16x16x4_f32 ----
    // 64 wave jobs: bit0 = re/im, bits1..3 = N tile (8), bits4..5 = M tile (4)
    const int wave = t >> 5;            // uniform within a wave -> EXEC stays full
    const int lane = t & 31;
    const int lo   = lane & 15;
    const int hi   = lane >> 4;

    for (int jb = wave; jb < 64; jb += 2) {
        const int part  = jb & 1;
        const int tileN = (jb >> 1) & 7;
        const int tileM = jb >> 4;
        const float (*X)[KP] = part ? Xim : Xre;
        float (*Y)[NC]       = part ? Yim : Yre;

        const int am = tileM * 16 + lo;     // A: M = lane%16
        const int bn = tileN * 16 + lo;     // B: N = lane%16

        v8f acc = {};
        #pragma unroll
        for (int ks = 0; ks < KP; ks += 4) {
            const int k0 = ks + hi * 2;     // VGPR0 -> K=k0, VGPR1 -> K=k0+1
            v2f a = { X[am][k0], X[am][k0 + 1] };
            v2f b = { Wl[bn][k0], Wl[bn][k0 + 1] };
            acc = __builtin_amdgcn_wmma_f32_16x16x4_f32(
                      false, a, false, b, (short)0, acc, false, false);
        }
        // C/D layout: VGPR r, lanes<16 -> M=r, lanes>=16 -> M=r+8; N=lane%16
        const int n     = tileN * 16 + lo;
        const int mbase = tileM * 16 + hi * 8;
        #pragma unroll
        for (int r = 0; r < 8; ++r) Y[mbase + r][n] = acc[r];
    }
    __syncthreads();

    // ---- tree phase (f64): one thread per batch row ----
    double nr[32], ni[32];

    #pragma unroll
    for (int j = 0; j < 32; ++j) {      // leaf level: pairs of Y cols 0..63
        double ar = (double)Yre[t][2 * j],     ai = (double)Yim[t][2 * j];
        double br = (double)Yre[t][2 * j + 1], bi = (double)Yim[t][2 * j + 1];
        eml_c(ar, ai, br, bi, nr[j], ni[j]);
    }

    int colbase = 64;                   // internal Y columns start at 64
    int cbase   = 0;
    #pragma unroll
    for (int n2 = 32; n2 >= 2; n2 >>= 1) {      // 2n = 32,16,8,4,2
        #pragma unroll
        for (int j = 0; j < 16; ++j) {
            if (j < (n2 >> 1)) {
                const int j0 = 2 * j, j1 = 2 * j + 1;
                double c0 = cl[cbase + j0], c1 = cl[cbase + j1];
                double ar = (double)Yre[t][colbase + j0] + nr[j0] * c0;
                double ai = (double)Yim[t][colbase + j0] + ni[j0] * c0;
                double br = (double)Yre[t][colbase + j1] + nr[j1] * c1;
                double bi = (double)Yim[t][colbase + j1] + ni[j1] * c1;
                eml_c(ar, ai, br, bi, nr[j], ni[j]);
            }
        }
        colbase += n2;
        cbase   += n2;
    }

    out[(size_t)(r0 + t) * 2]     = nr[0];
    out[(size_t)(r0 + t) * 2 + 1] = ni[0];
}

// ---------------------------------------------------------------------------
// Launch: softmax weights -> fused WMMA GEMM + f64 tree.
// d_in[0] = x (complex128 -> interleaved double), d_in[1] = internal_logits,
// d_in[2] = leaf_logits. d_out = (B,) complex128 (interleaved double).
// ---------------------------------------------------------------------------
extern "C" void kernel_launch(void* const* d_in, const int* in_sizes, int n_in,
                              void* d_out, int out_size, void* d_ws, size_t ws_size,
                              hipStream_t stream)
{
    (void)in_sizes; (void)n_in; (void)out_size; (void)ws_size;

    const double* x        = (const double*)d_in[0];
    const double* internal = (const double*)d_in[1];
    const double* leaf     = (const double*)d_in[2];
    double*       out      = (double*)d_out;

    float*  W  = (float*)d_ws;                                   // NC*KP f32
    double* cv = (double*)((char*)d_ws + NC * KP * sizeof(float)); // 62 f64

    hipLaunchKernelGGL(eml_weights_kernel, dim3(1), dim3(128), 0, stream,
                       internal, leaf, W, cv);

    hipLaunchKernelGGL(eml_fused_kernel, dim3(BATCH / ROWS), dim3(THREADS), 0, stream,
                       x, W, cv, out);
}